// AWGNIndexChannelWrapper_72052371358246
// MI455X (gfx1250) — compile-verified
//
#include <hip/hip_runtime.h>

// AWGN index channel + codebook gather for MI455X (gfx1250, wave32).
//
// Reference semantics:
//   flip bit j of idx iff u[j] < 0.02 (j = 0..8, LSB-first), clip to [0,511],
//   gather codebook[rx] (128 floats), concat [fine | coarse] per batch row.
//
// Mapping: one wave32 per index. Lanes 0..8 test their uniform < BER; a wave
// ballot's low 9 bits are exactly the XOR flip mask (wave-uniform in an SGPR).
// Then all 32 lanes copy the 512-byte codebook row with one v4f (b128)
// load + one nontemporal v4f store each. Write-dominated: ~168 MB out,
// ~13 MB in; codebooks (512 KB) stay L2-resident for the gathers.
//
// wmma is intentionally absent: this workload has zero matrix-multiply FLOPs.

typedef float v4f __attribute__((ext_vector_type(4)));   // native vec for NT builtin

#define BER   0.02f
#define DDIM  128
// per-batch output row stride in floats: 64*64*128 (fine) + 32*32*128 (coarse)
#define OUT_STRIDE 655360LL
#define FINE_FLOATS_PER_B 524288LL   // 64*64*128

template <int HW_SHIFT>   // H*W per batch = 1 << HW_SHIFT (power of two)
__global__ __launch_bounds__(256) void bitflip_gather_kernel(
    const int*   __restrict__ idx_arr,
    const float* __restrict__ flip_u,
    const float* __restrict__ codebook,
    float*       __restrict__ out,
    int n_idx,            // total number of indices (B * H * W)
    long long base_off)   // float offset of this section inside a batch row
{
    const int lane   = threadIdx.x & 31;
    const int waveId = (int)((blockIdx.x * blockDim.x + threadIdx.x) >> 5);
    const int nWaves = (int)((gridDim.x * blockDim.x) >> 5);

    for (int i = waveId; i < n_idx; i += nWaves) {
        // --- 9-bit bit-flip channel via wave32 ballot ---------------------
        float u = 1.0f;
        if (lane < 9) u = flip_u[(size_t)i * 9 + lane];
        unsigned mask = __builtin_amdgcn_ballot_w32(u < BER) & 0x1FFu;

        int v  = idx_arr[i];                 // wave-uniform address
        int rx = v ^ (int)mask;              // stays in [0,512) for 9 bits
        rx = rx < 511 ? rx : 511;            // clip (matches reference)

        // --- copy one codebook row: 32 lanes x v4f = 128 floats -----------
        const v4f val =
            ((const v4f*)(codebook + (size_t)rx * DDIM))[lane];

        const int b   = i >> HW_SHIFT;
        const int pos = i & ((1 << HW_SHIFT) - 1);
        v4f* dst = (v4f*)(out + (long long)b * OUT_STRIDE + base_off
                              + (long long)pos * DDIM);
        __builtin_nontemporal_store(val, dst + lane);
    }
}

extern "C" void kernel_launch(void* const* d_in, const int* in_sizes, int n_in,
                              void* d_out, int out_size, void* d_ws, size_t ws_size,
                              hipStream_t stream) {
    (void)in_sizes; (void)n_in; (void)out_size; (void)d_ws; (void)ws_size;

    const int*   idx_c = (const int*)  d_in[0];   // [64,32,32] int32
    const int*   idx_f = (const int*)  d_in[1];   // [64,64,64] int32
    const float* cb_c  = (const float*)d_in[2];   // [512,128]  f32
    const float* cb_f  = (const float*)d_in[3];   // [512,128]  f32
    const float* u_c   = (const float*)d_in[4];   // [64,32,32,9] f32
    const float* u_f   = (const float*)d_in[5];   // [64,64,64,9] f32
    float* out = (float*)d_out;                   // [64, 655360] f32

    const int Nf = 64 * 64 * 64;   // 262144 fine indices
    const int Nc = 64 * 32 * 32;   //  65536 coarse indices

    // Fine section first (offset 0 in each batch row), then coarse.
    bitflip_gather_kernel<12><<<4096, 256, 0, stream>>>(
        idx_f, u_f, cb_f, out, Nf, 0LL);
    bitflip_gather_kernel<10><<<1024, 256, 0, stream>>>(
        idx_c, u_c, cb_c, out, Nc, FINE_FLOATS_PER_B);
}